// GraphSage_43293270344193
// MI455X (gfx1250) — compile-verified
//
#include <hip/hip_runtime.h>
#include <hip/hip_bf16.h>

typedef __attribute__((ext_vector_type(2))) float v2f;
typedef __attribute__((ext_vector_type(8))) float v8f;

// Match the builtin's parameter types exactly: vector_size(16) int, AS1/AS3.
typedef int v4i_ __attribute__((vector_size(16)));
typedef __attribute__((address_space(1))) v4i_* as1_v4i;
typedef __attribute__((address_space(3))) v4i_* as3_v4i;

#define DN_   256
#define DE_   128
#define DIN_  640
#define DOUT_ 256
#define BLK_M 128
#define BLK_N 128
#define TILE_K 16
#define LDS_STRIDE 20   // 16 + 4 pad; rows stay 16B-aligned (80B) and bank-spread

// ---------------------------------------------------------------------------
// Async global -> LDS copy of 16 bytes (CDNA5 GLOBAL_LOAD_ASYNC_TO_LDS_B128,
// tracked by ASYNCcnt). Builtin if present, else raw ISA per CDNA5 ISA 15.18.
// ---------------------------------------------------------------------------
__device__ __forceinline__ void async_copy_b128(const float* g, float* l) {
#if __has_builtin(__builtin_amdgcn_global_load_async_to_lds_b128)
    __builtin_amdgcn_global_load_async_to_lds_b128(
        (as1_v4i)g, (as3_v4i)l, /*imm offset=*/0, /*cpol=*/0);
#else
    asm volatile("global_load_async_to_lds_b128 %0, %1, off"
                 :
                 : "v"((unsigned)(unsigned long long)
                           (__attribute__((address_space(3))) float*)l),
                   "v"(g)
                 : "memory");
#endif
}

template <int N>
__device__ __forceinline__ void wait_asynccnt() {
#if __has_builtin(__builtin_amdgcn_s_wait_asynccnt)
    __builtin_amdgcn_s_wait_asynccnt(N);
#else
    asm volatile("s_wait_asynccnt %0" ::"n"(N) : "memory");
#endif
}

// ---------------------------------------------------------------------------
// combined[m, 0:256)  = src_feats[self_idx[m], :]   (gather)
// combined[m, 256:640) = 0                          (atomic accumulation target)
// ---------------------------------------------------------------------------
__global__ __launch_bounds__(256) void init_combined(
    const float* __restrict__ src_feats,   // [*, 256]
    const int*   __restrict__ self_idx,    // [M]
    float*       __restrict__ combined,    // [M, 640]
    int M)
{
    long tid = (long)blockIdx.x * blockDim.x + threadIdx.x;
    int m = (int)(tid / DIN_);
    int c = (int)(tid % DIN_);
    if (m >= M) return;
    float v = 0.0f;
    if (c < DN_) v = src_feats[(size_t)self_idx[m] * DN_ + c];
    combined[(size_t)m * DIN_ + c] = v;
}

// ---------------------------------------------------------------------------
// For each edge e: combined[seg_idx[e], 256 + c] += (c<256 ? nf[from[e],c]
//                                                          : ef[e, c-256])
// One block (384 threads) per edge: coalesced row gather + HW fp32 atomics.
// ---------------------------------------------------------------------------
__global__ __launch_bounds__(384) void scatter_edges(
    const float* __restrict__ nf,        // [N, 256]
    const float* __restrict__ ef,        // [E, 128]
    const int*   __restrict__ from_idx,  // [E]
    const int*   __restrict__ seg_idx,   // [E]
    float*       __restrict__ combined,  // [M, 640]; agg region at col 256
    int E)
{
    int e = blockIdx.x;
    if (e >= E) return;
    int c = threadIdx.x;                 // 0..383
    float v = (c < DN_) ? nf[(size_t)from_idx[e] * DN_ + c]
                        : ef[(size_t)e * DE_ + (c - DN_)];
    unsafeAtomicAdd(&combined[(size_t)seg_idx[e] * DIN_ + DN_ + c], v);
}

// ---------------------------------------------------------------------------
// Out = relu(A @ W^T), A:[M,640] fp32, W:[256,640] fp32, Out:[M,256].
// Block: 256 threads = 8 waves -> 128x128 output tile; wave w owns rows
// [w*16, w*16+16) x eight 16x16 sub-tiles. K loop uses V_WMMA_F32_16X16X4_F32.
// Tiles are staged with async global->LDS copies, double-buffered:
//   issue(k+1 -> buf^1); s_wait_asynccnt 4 (tile k landed); barrier; compute.
// ---------------------------------------------------------------------------
__global__ __launch_bounds__(256) void sage_gemm_relu(
    const float* __restrict__ A,    // [M, 640]
    const float* __restrict__ W,    // [256, 640]
    float*       __restrict__ Out,  // [M, 256]
    int M)
{
    __shared__ float lA[2][BLK_M * LDS_STRIDE];
    __shared__ float lB[2][BLK_N * LDS_STRIDE];

    const int tid    = threadIdx.x;
    const int wave   = tid >> 5;          // 0..7
    const int lane   = tid & 31;
    const int lm     = lane & 15;         // m (A) / n (B,D) within tile
    const int kh     = (lane >> 4) << 1;  // K half-offset: 0 or 2
    const int rowBlk = blockIdx.x * BLK_M;
    const int colBlk = blockIdx.y * BLK_N;

    // Cooperative tile mapping: 256 threads x 8 floats = 128x16 tile (x2 b128).
    const int ldRow = tid >> 1;           // 0..127
    const int ldCol = (tid & 1) << 3;     // 0 or 8

    const float* gA = A + (size_t)(rowBlk + ldRow) * DIN_ + ldCol;
    const float* gW = W + (size_t)(colBlk + ldRow) * DIN_ + ldCol;
    float* dA[2] = { &lA[0][ldRow * LDS_STRIDE + ldCol],
                     &lA[1][ldRow * LDS_STRIDE + ldCol] };
    float* dB[2] = { &lB[0][ldRow * LDS_STRIDE + ldCol],
                     &lB[1][ldRow * LDS_STRIDE + ldCol] };

    v8f acc[8] = {};

    auto issue = [&](int k0, int buf) {
        async_copy_b128(gA + k0,     dA[buf]);
        async_copy_b128(gA + k0 + 4, dA[buf] + 4);
        async_copy_b128(gW + k0,     dB[buf]);
        async_copy_b128(gW + k0 + 4, dB[buf] + 4);
    };

    auto compute = [&](const float* bufA, const float* bufB) {
#pragma unroll
        for (int kk = 0; kk < TILE_K; kk += 4) {
            const float* ap = &bufA[(wave * 16 + lm) * LDS_STRIDE + kk + kh];
            v2f afrag;
            afrag.x = ap[0];
            afrag.y = ap[1];
#pragma unroll
            for (int t = 0; t < 8; ++t) {
                const float* bp = &bufB[(t * 16 + lm) * LDS_STRIDE + kk + kh];
                v2f bfrag;
                bfrag.x = bp[0];
                bfrag.y = bp[1];
                acc[t] = __builtin_amdgcn_wmma_f32_16x16x4_f32(
                    /*neg_a=*/false, afrag,
                    /*neg_b=*/false, bfrag,
                    /*c_mod=*/(short)0, acc[t],
                    /*reuse_a=*/false, /*reuse_b=*/false);
            }
        }
    };

    const int nt = DIN_ / TILE_K;  // 40 K-tiles
    issue(0, 0);
    for (int k = 0; k < nt; ++k) {
        const int buf = k & 1;
        if (k + 1 < nt) {
            // Prefetch next tile into the other buffer (consumed last iter,
            // protected by the trailing barrier), then wait for tile k's 4
            // in-order async ops to land (<=4 leaves the new 4 in flight).
            issue((k + 1) * TILE_K, buf ^ 1);
            wait_asynccnt<4>();
        } else {
            wait_asynccnt<0>();
        }
        __syncthreads();
        compute(&lA[buf][0], &lB[buf][0]);
        __syncthreads();
    }

    // ReLU + store. VGPR r of D holds row (r + 8*(lane>=16)), col = lane&15.
    const int outRow0 = rowBlk + wave * 16 + ((lane >> 4) << 3);
#pragma unroll
    for (int t = 0; t < 8; ++t) {
        const int outCol = colBlk + t * 16 + lm;
#pragma unroll
        for (int r = 0; r < 8; ++r) {
            float v = acc[t][r];
            v = v > 0.0f ? v : 0.0f;
            Out[(size_t)(outRow0 + r) * DOUT_ + outCol] = v;
        }
    }
}

// ---------------------------------------------------------------------------
extern "C" void kernel_launch(void* const* d_in, const int* in_sizes, int n_in,
                              void* d_out, int out_size, void* d_ws, size_t ws_size,
                              hipStream_t stream) {
    const float* nf    = (const float*)d_in[0];   // [N, 256]
    const float* ef1   = (const float*)d_in[1];   // [E1, 128]
    const float* ef2   = (const float*)d_in[2];   // [E2, 128]
    const float* W1    = (const float*)d_in[3];   // [256, 640]
    const float* W2    = (const float*)d_in[4];   // [256, 640]
    const int*   from1 = (const int*)d_in[5];     // [E1]
    const int*   seg1  = (const int*)d_in[6];     // [E1]
    const int*   self1 = (const int*)d_in[7];     // [M1]
    const int*   from2 = (const int*)d_in[8];     // [E2]
    const int*   seg2  = (const int*)d_in[9];     // [E2]
    const int*   self2 = (const int*)d_in[10];    // [B]

    const int E1 = in_sizes[5];
    const int M1 = in_sizes[7];
    const int E2 = in_sizes[8];
    const int Bv = in_sizes[10];

    float* out = (float*)d_out;

    // Workspace layout (fp32):
    //   combined1 : [M1, 640]
    //   h1        : [M1, 256]
    //   combined2 : [B,  640]
    float* combined1 = (float*)d_ws;
    float* h1        = combined1 + (size_t)M1 * DIN_;
    float* combined2 = h1 + (size_t)M1 * DOUT_;

    // ---- Layer 1 ----
    {
        long total = (long)M1 * DIN_;
        int blocks = (int)((total + 255) / 256);
        init_combined<<<blocks, 256, 0, stream>>>(nf, self1, combined1, M1);
    }
    scatter_edges<<<E1, 384, 0, stream>>>(nf, ef1, from1, seg1, combined1, E1);
    {
        dim3 grid(M1 / BLK_M, DOUT_ / BLK_N);
        sage_gemm_relu<<<grid, 256, 0, stream>>>(combined1, W1, h1, M1);
    }

    // ---- Layer 2 ----
    {
        long total = (long)Bv * DIN_;
        int blocks = (int)((total + 255) / 256);
        init_combined<<<blocks, 256, 0, stream>>>(h1, self2, combined2, Bv);
    }
    scatter_edges<<<E2, 384, 0, stream>>>(nf, ef2, from2, seg2, combined2, E2);
    {
        dim3 grid(Bv / BLK_M, DOUT_ / BLK_N);
        sage_gemm_relu<<<grid, 256, 0, stream>>>(combined2, W2, out, Bv);
    }
}